// MinimumErrorRateLoss_3556232921404
// MI455X (gfx1250) — compile-verified
//
#include <hip/hip_runtime.h>
#include <stdint.h>

typedef __attribute__((ext_vector_type(2))) float v2f;
typedef __attribute__((ext_vector_type(8))) float v8f;

#define R_TOK  256
#define H_TOK  256
#define NBATCH 128
#define NSAMP  64
#define NSEQ   (NBATCH * NSAMP)   /* 8192 sequences */

// ---------------------------------------------------------------------------
// Kernel 1: one wave32 per sequence. Levenshtein DP over H hyp steps; the
// deletion chain along the ref axis is a min-plus prefix scan:
//   new_row[j] = j + cummin_{k<=j}(cand[k] - k),  cand[0] = i.
// Each lane owns 8 consecutive row entries (j = 8*lane+1 .. 8*lane+8).
// Hyp tokens are staged to LDS with gfx1250 async global->LDS loads.
// ---------------------------------------------------------------------------
__global__ __launch_bounds__(256) void mer_edit_distance_kernel(
    const int* __restrict__ ref,   // (256,128) int32, row-major
    const int* __restrict__ hyp,   // (256,128,64) int32 -> (256, 8192)
    float* __restrict__ er)        // (8192,) error rates out
{
  __shared__ int hypbuf[8][H_TOK];          // 1 KB per wave, 8 KB per block
  const int tid  = threadIdx.x;
  const int w    = tid >> 5;
  const int lane = tid & 31;
  const int n    = blockIdx.x * 8 + w;      // sequence id 0..8191
  const int b    = n >> 6;                  // batch row (ref column)

  int* hypw = hypbuf[w];

  // ---- async-stage this sequence's 256 hyp tokens into LDS (ASYNCcnt path)
  #pragma unroll
  for (int t = 0; t < 8; ++t) {
    const int i0 = lane + 32 * t;
    const int* gsrc = hyp + (size_t)i0 * NSEQ + n;
    unsigned int       lds_off = (unsigned int)(uintptr_t)(&hypw[i0]);
    unsigned long long gaddr   = (unsigned long long)(uintptr_t)gsrc;
    asm volatile("global_load_async_to_lds_b32 %0, %1, off"
                 :: "v"(lds_off), "v"(gaddr) : "memory");
  }

  // ---- ref tokens for my 8 DP cells (reused all 256 steps, stay in VGPRs)
  int rtok[8];
  #pragma unroll
  for (int t = 0; t < 8; ++t)
    rtok[t] = ref[(8 * lane + t) * NBATCH + b];

  // ref_len = first EOS index + 1, else 256 (wave min-reduce)
  int rfirst = 0x3FFFFFFF;
  #pragma unroll
  for (int t = 0; t < 8; ++t)
    if (rtok[t] == 0) rfirst = min(rfirst, 8 * lane + t);
  #pragma unroll
  for (int d = 1; d < 32; d <<= 1)
    rfirst = min(rfirst, __shfl_xor(rfirst, d, 32));
  const int ref_len = (rfirst < R_TOK) ? rfirst + 1 : R_TOK;

  // wait for the async LDS staging of this wave
  asm volatile("s_wait_asynccnt 0" ::: "memory");

  // hyp_len (same rule) from the staged tokens
  int hfirst = 0x3FFFFFFF;
  #pragma unroll
  for (int t = 0; t < 8; ++t) {
    const int v = hypw[8 * lane + t];
    if (v == 0) hfirst = min(hfirst, 8 * lane + t);
  }
  #pragma unroll
  for (int d = 1; d < 32; d <<= 1)
    hfirst = min(hfirst, __shfl_xor(hfirst, d, 32));
  const int hyp_len = (hfirst < H_TOK) ? hfirst + 1 : H_TOK;

  // ---- DP. Rows freeze after i > hyp_len, so simply stop there.
  const int jbase = 8 * lane + 1;
  int row[8];
  #pragma unroll
  for (int t = 0; t < 8; ++t) row[t] = jbase + t;   // row0[j] = j

  for (int i = 1; i <= hyp_len; ++i) {
    const int h = hypw[i - 1];                       // uniform broadcast read
    int left = __shfl_up(row[7], 1, 32);             // row_old[8*lane]
    if (lane == 0) left = i - 1;                     // row_old[0]

    int m[8];
    int diag = left;
    #pragma unroll
    for (int t = 0; t < 8; ++t) {                    // cand = min(up+1, diag+sub)
      const int cand = min(row[t] + 1, diag + (rtok[t] != h ? 1 : 0));
      diag = row[t];
      m[t] = cand - (jbase + t);                     // min-plus transform
    }

    int tot = m[0];
    #pragma unroll
    for (int t = 1; t < 8; ++t) tot = min(tot, m[t]);
    #pragma unroll
    for (int d = 1; d < 32; d <<= 1) {               // inclusive wave min-scan
      const int y = __shfl_up(tot, d, 32);
      if (lane >= d) tot = min(tot, y);
    }
    int pre = __shfl_up(tot, 1, 32);                 // exclusive prefix
    if (lane == 0) pre = 0x3FFFFFFF;
    pre = min(pre, i);                               // cand[0] - 0 = i

    int run = pre;
    #pragma unroll
    for (int t = 0; t < 8; ++t) {
      run = min(run, m[t]);
      row[t] = run + jbase + t;
    }
  }

  // park the final row in LDS (hyp buffer is dead now) and pick row[ref_len]
  #pragma unroll
  for (int t = 0; t < 8; ++t) hypw[jbase + t - 1] = row[t];
  const int dist = hypw[ref_len - 1];                // in-order DS, same wave
  if (lane == 0) er[n] = (float)dist / (float)ref_len;
}

// extract element k (runtime) of an 8-wide f32 accumulator, branchless
__device__ __forceinline__ float pick8(v8f a, int k) {
  float r = a[0];
  #pragma unroll
  for (int q = 1; q < 8; ++q) r = (k == q) ? a[q] : r;
  return r;
}

// ---------------------------------------------------------------------------
// Kernel 2: scalar loss. 8 waves, one 16-row tile of the (128,64) matrices
// per wave. All three per-row reductions (softmax denominator Σexp, Σer,
// Σ er*exp) are 16x64 row-sums = WMMA against an all-ones B matrix, chained
// over K in 16 steps of v_wmma_f32_16x16x4_f32.
//   loss = (1/8192) * Σ_b [ Σ(er·e)/Σe − (Σer)/64 ]
// ---------------------------------------------------------------------------
__global__ __launch_bounds__(256) void mer_loss_kernel(
    const float* __restrict__ lp,   // (128,64) log-probs
    const float* __restrict__ er,   // (128,64) error rates (from d_ws)
    float* __restrict__ out)        // scalar
{
  __shared__ float wred[8];
  const int tid  = threadIdx.x;
  const int w    = tid >> 5;
  const int lane = tid & 31;
  const int mrow = w * 16 + (lane & 15);        // my matrix row
  const int koff = (lane < 16) ? 0 : 2;         // A 16x4 layout: K split 0/1 | 2/3

  // pass 1: row max for a stable softmax (lane pair <->lane^16 covers a row)
  float mx = -3.0e38f;
  for (int kk = 0; kk < 16; ++kk) {
    const int c0 = 4 * kk + koff;
    mx = fmaxf(mx, fmaxf(lp[mrow * 64 + c0], lp[mrow * 64 + c0 + 1]));
  }
  mx = fmaxf(mx, __shfl_xor(mx, 16, 32));

  // pass 2: three chained WMMA row-sum accumulators
  v2f ones; ones.x = 1.0f; ones.y = 1.0f;
  v8f accD  = {0.f, 0.f, 0.f, 0.f, 0.f, 0.f, 0.f, 0.f};
  v8f accER = {0.f, 0.f, 0.f, 0.f, 0.f, 0.f, 0.f, 0.f};
  v8f accEE = {0.f, 0.f, 0.f, 0.f, 0.f, 0.f, 0.f, 0.f};
  for (int kk = 0; kk < 16; ++kk) {
    const int c0 = 4 * kk + koff;
    const float e0 = __expf(lp[mrow * 64 + c0]     - mx);
    const float e1 = __expf(lp[mrow * 64 + c0 + 1] - mx);
    const float r0 = er[mrow * 64 + c0];
    const float r1 = er[mrow * 64 + c0 + 1];
    v2f aE; aE.x = e0;       aE.y = e1;
    v2f aR; aR.x = r0;       aR.y = r1;
    v2f aP; aP.x = r0 * e0;  aP.y = r1 * e1;
    accD  = __builtin_amdgcn_wmma_f32_16x16x4_f32(false, aE, false, ones,
                                                  (short)0, accD,  false, false);
    accER = __builtin_amdgcn_wmma_f32_16x16x4_f32(false, aR, false, ones,
                                                  (short)0, accER, false, false);
    accEE = __builtin_amdgcn_wmma_f32_16x16x4_f32(false, aP, false, ones,
                                                  (short)0, accEE, false, false);
  }

  // D layout: vgpr q = row q (lanes 0-15) / row q+8 (lanes 16-31); all columns
  // of D equal the row sum (B == ones). Lane picks vgpr (lane&7): every row
  // appears exactly twice across the wave.
  const int k = lane & 7;
  const float sD  = pick8(accD,  k);
  const float sER = pick8(accER, k);
  const float sEE = pick8(accEE, k);
  float contrib = sEE / sD - sER * (1.0f / 64.0f);
  #pragma unroll
  for (int d = 1; d < 32; d <<= 1) contrib += __shfl_xor(contrib, d, 32);
  if (lane == 0) wred[w] = contrib;              // = 2 * Σ_rows-in-tile
  __syncthreads();
  if (tid == 0) {
    float tot = 0.f;
    for (int q = 0; q < 8; ++q) tot += wred[q];
    out[0] = tot / (2.0f * (float)NSEQ);         // undo x2, then mean over 8192
  }
}

extern "C" void kernel_launch(void* const* d_in, const int* in_sizes, int n_in,
                              void* d_out, int out_size, void* d_ws, size_t ws_size,
                              hipStream_t stream) {
  (void)in_sizes; (void)n_in; (void)out_size; (void)ws_size;
  const float* lp  = (const float*)d_in[0];   // (128,64) f32
  const int*   ref = (const int*)d_in[1];     // (256,128) int
  const int*   hyp = (const int*)d_in[2];     // (256,128,64) int
  float* er_ws = (float*)d_ws;                // 8192 floats of scratch
  float* out   = (float*)d_out;

  mer_edit_distance_kernel<<<NSEQ / 8, 256, 0, stream>>>(ref, hyp, er_ws);
  mer_loss_kernel<<<1, 256, 0, stream>>>(lp, er_ws, out);
}